// GCN_89163521065703
// MI455X (gfx1250) — compile-verified
//
#include <hip/hip_runtime.h>

typedef __attribute__((ext_vector_type(2))) float v2f;
typedef __attribute__((ext_vector_type(8))) float v8f;

#define NNODES 50000
#define NEDGES 800000
#define DIN_   128
#define DH_    256
#define DOUT_  40

// ---- degree / normalization -------------------------------------------------

__global__ void k_deg_init(float* deg, int n) {
    int i = blockIdx.x * blockDim.x + threadIdx.x;
    if (i < n) deg[i] = 1.0f;                  // self-loop contributes 1
}

__global__ void k_deg_accum(const int* __restrict__ dst, float* deg, int e) {
    int i = blockIdx.x * blockDim.x + threadIdx.x;
    if (i < e) atomicAdd(&deg[dst[i]], 1.0f);
}

__global__ void k_dinv(float* deg, int n) {
    int i = blockIdx.x * blockDim.x + threadIdx.x;
    if (i < n) {
        float d = deg[i];
        deg[i] = (d > 0.f) ? rsqrtf(d) : 0.f;
    }
}

// ---- WMMA f32 GEMM ----------------------------------------------------------
// D[M,NV] = A[M,K] * B[K,NV] (all row-major, ld = K / NV / NV).
// One wave computes a 64x16 tile: 4 M sub-tiles sharing each B fragment,
// 4 independent accumulator chains. V_WMMA_F32_16X16X4_F32, K stepped by 4.
// COLCHK=false when gridDim.y*16 == NV (no column predication at all).

template<int K, int NV, bool COLCHK>
__global__ void k_gemm_wmma(const float* __restrict__ A, const float* __restrict__ B,
                            float* __restrict__ D, int M) {
    const int mt = blockIdx.x * blockDim.y + threadIdx.y;   // 64-row tile index
    const int m0 = mt * 64;
    if (m0 >= M) return;                                    // wave-uniform exit

    const int lane = threadIdx.x & 31;
    const int half = lane >> 4;                             // 0: lanes 0-15, 1: 16-31
    const int lm   = lane & 15;
    const int col  = blockIdx.y * 16 + lm;
    const bool colok = !COLCHK || (col < NV);
    const int  colc  = colok ? col : (NV - 1);              // clamped, always in-bounds

    // Per-sub-tile A row pointers (clamped to M-16 for the partial last tile;
    // M is a multiple of 16 so clamped sub-tiles are exactly the skipped ones).
    int mbase[4];
    const float* ar[4];
#pragma unroll
    for (int s = 0; s < 4; ++s) {
        int mb = m0 + 16 * s;
        if (mb + 16 > M) mb = M - 16;
        mbase[s] = mb;
        // lane lm holds row mb+lm; a.x = K=kk+2*half, a.y = K=kk+2*half+1
        ar[s] = A + (size_t)(mb + lm) * K + 2 * half;
    }
    // B: lane lm holds col; b.x = row kk+2*half, b.y = row kk+2*half+1
    const float* bp = B + (size_t)(2 * half) * NV + colc;

    v8f c[4] = {};
#pragma unroll 4
    for (int kk = 0; kk < K; kk += 4) {
        float bx = bp[(size_t)kk * NV];          // immediate offsets (K, NV constexpr)
        float by = bp[(size_t)kk * NV + NV];
        v2f b;
        b.x = colok ? bx : 0.f;                  // v_cndmask, no exec manipulation
        b.y = colok ? by : 0.f;
#pragma unroll
        for (int s = 0; s < 4; ++s) {
            v2f a = *(const v2f*)(ar[s] + kk);   // contiguous 8B -> global_load_b64
            c[s] = __builtin_amdgcn_wmma_f32_16x16x4_f32(false, a, false, b,
                                                         (short)0, c[s], false, false);
        }
    }

#pragma unroll
    for (int s = 0; s < 4; ++s) {
        if (colok && (m0 + 16 * s) < M) {
#pragma unroll
            for (int i = 0; i < 8; ++i)  // VGPR i: rows mb+i (lanes 0-15), mb+8+i (16-31)
                D[(size_t)(mbase[s] + i + 8 * half) * NV + col] = c[s][i];
        }
    }
}

// ---- aggregation ------------------------------------------------------------

// out[i][c] = h[i][c] * dinv[i]^2   (self-loop term; fully initializes out)
__global__ void k_self_init(const float* __restrict__ h, const float* __restrict__ dinv,
                            float* __restrict__ out, int total4, int c4) {
    int t = blockIdx.x * blockDim.x + threadIdx.x;
    if (t >= total4) return;
    int node = t / c4;
    float s = dinv[node]; s *= s;
    float4 v = ((const float4*)h)[t];
    v.x *= s; v.y *= s; v.z *= s; v.w *= s;
    ((float4*)out)[t] = v;
}

// out[dst][c] += h[src][c] * dinv[src]*dinv[dst]; thread = (edge, 4-channel chunk)
__global__ void k_edge_scatter(const int* __restrict__ src, const int* __restrict__ dst,
                               const float* __restrict__ dinv, const float* __restrict__ h,
                               float* __restrict__ out, int e, int c4, int C) {
    int t = blockIdx.x * blockDim.x + threadIdx.x;
    if (t >= e * c4) return;
    int ed = t / c4;
    int cc = t % c4;            // consecutive lanes -> consecutive channels (coalesced gather)
    int s = src[ed], d = dst[ed];
    float norm = dinv[s] * dinv[d];
    float4 v = *(const float4*)(h + (size_t)s * C + cc * 4);
    float* op = out + (size_t)d * C + cc * 4;
    atomicAdd(op + 0, v.x * norm);
    atomicAdd(op + 1, v.y * norm);
    atomicAdd(op + 2, v.z * norm);
    atomicAdd(op + 3, v.w * norm);
}

__global__ void k_bias_relu(float* __restrict__ a, const float* __restrict__ b,
                            int total, int C) {
    int t = blockIdx.x * blockDim.x + threadIdx.x;
    if (t < total) {
        float v = a[t] + b[t % C];
        a[t] = (v > 0.f) ? v : 0.f;
    }
}

// in-place log_softmax over rows of DOUT_, with bias add fused
__global__ void k_logsoftmax(float* __restrict__ o, const float* __restrict__ b2, int n) {
    int i = blockIdx.x * blockDim.x + threadIdx.x;
    if (i >= n) return;
    float* row = o + (size_t)i * DOUT_;
    float m = -3.402823466e38f;
    for (int c = 0; c < DOUT_; ++c) {
        float v = row[c] + b2[c];
        m = fmaxf(m, v);
    }
    float ssum = 0.f;
    for (int c = 0; c < DOUT_; ++c) ssum += expf(row[c] + b2[c] - m);
    float l = logf(ssum);
    for (int c = 0; c < DOUT_; ++c) row[c] = row[c] + b2[c] - m - l;
}

// ---- launch -----------------------------------------------------------------

extern "C" void kernel_launch(void* const* d_in, const int* in_sizes, int n_in,
                              void* d_out, int out_size, void* d_ws, size_t ws_size,
                              hipStream_t stream) {
    const float* x  = (const float*)d_in[0];
    const int*   ei = (const int*)  d_in[1];   // [2, E]
    const float* W1 = (const float*)d_in[2];   // [128, 256]
    const float* b1 = (const float*)d_in[3];
    const float* W2 = (const float*)d_in[4];   // [256, 40]
    const float* b2 = (const float*)d_in[5];
    const int* src = ei;
    const int* dst = ei + NEDGES;

    float* ws   = (float*)d_ws;
    float* dinv = ws;                                 // 50000 (deg -> dinv in place)
    float* h1   = ws + 50048;                         // 50000*256 (reused for h2)
    float* a1   = h1 + (size_t)NNODES * DH_;          // 50000*256
    float* h2   = h1;                                 // h1 dead after layer-1 agg
    float* outp = (float*)d_out;                      // 50000*40, doubles as agg2

    dim3 b256(256);

    // normalization
    k_deg_init <<<(NNODES + 255) / 256, b256, 0, stream>>>(dinv, NNODES);
    k_deg_accum<<<(NEDGES + 255) / 256, b256, 0, stream>>>(dst, dinv, NEDGES);
    k_dinv     <<<(NNODES + 255) / 256, b256, 0, stream>>>(dinv, NNODES);

    // layer 1: h1 = x @ W1.  64-row tiles: ceil(50000/64)=782 waves along M.
    dim3 gblk(32, 4);                                 // 4 waves/block, 256 rows/block
    {
        int mtiles = (NNODES + 63) / 64;              // 782
        dim3 grid((mtiles + 3) / 4, DH_ / 16);        // (196, 16)
        k_gemm_wmma<DIN_, DH_, false><<<grid, gblk, 0, stream>>>(x, W1, h1, NNODES);
    }

    int t4 = NNODES * (DH_ / 4);
    k_self_init<<<(t4 + 255) / 256, b256, 0, stream>>>(h1, dinv, a1, t4, DH_ / 4);
    int te = NEDGES * (DH_ / 4);
    k_edge_scatter<<<(te + 255) / 256, b256, 0, stream>>>(
        src, dst, dinv, h1, a1, NEDGES, DH_ / 4, DH_);
    k_bias_relu<<<(NNODES * DH_ + 255) / 256, b256, 0, stream>>>(
        a1, b1, NNODES * DH_, DH_);

    // layer 2: h2 = a1 @ W2  (N=40 -> 3 column tiles, predicated)
    {
        int mtiles = (NNODES + 63) / 64;
        dim3 grid((mtiles + 3) / 4, 3);               // (196, 3) covers 48 cols
        k_gemm_wmma<DH_, DOUT_, true><<<grid, gblk, 0, stream>>>(a1, W2, h2, NNODES);
    }

    int t4b = NNODES * (DOUT_ / 4);
    k_self_init<<<(t4b + 255) / 256, b256, 0, stream>>>(h2, dinv, outp, t4b, DOUT_ / 4);
    int teb = NEDGES * (DOUT_ / 4);
    k_edge_scatter<<<(teb + 255) / 256, b256, 0, stream>>>(
        src, dst, dinv, h2, outp, NEDGES, DOUT_ / 4, DOUT_);

    k_logsoftmax<<<(NNODES + 255) / 256, b256, 0, stream>>>(outp, b2, NNODES);
}